// Model_52209622450232
// MI455X (gfx1250) — compile-verified
//
#include <hip/hip_runtime.h>
#include <math.h>

// ---------------------------------------------------------------------------
// Problem constants (from the reference)
// ---------------------------------------------------------------------------
#define GZ   384          // grid size
#define HZ   192          // conv1 output size (stride 2, pad 2, k=5)
#define NCH  8            // conv channel count
#define NB   4            // batch
#define NPTS 200000       // particles per batch

typedef __attribute__((ext_vector_type(2))) float v2f;
typedef __attribute__((ext_vector_type(8))) float v8f;
typedef __attribute__((ext_vector_type(4))) int   v4i;

__device__ __forceinline__ float elu_f(float x)  { return x > 0.f ? x : (expf(x) - 1.f); }
__device__ __forceinline__ float elup_f(float x) { return x > 0.f ? 1.f : expf(x); }
__device__ __forceinline__ int   clampi(int v, int lo, int hi) {
  return v < lo ? lo : (v > hi ? hi : v);
}

// ---------------------------------------------------------------------------
// Async global->LDS staging (gfx1250 ASYNCcnt path), with safe fallback.
// Builtin signature (from hipcc diagnostic): first param is a pointer to
// int __vector(4) in the global address space; pass typed AS-qualified ptrs.
// ---------------------------------------------------------------------------
#if defined(__has_builtin)
#if __has_builtin(__builtin_amdgcn_global_load_async_to_lds_b128) && \
    __has_builtin(__builtin_amdgcn_s_wait_asynccnt)
#define ASYNC_LDS 1
#endif
#endif

typedef __attribute__((address_space(1))) v4i* as1_v4i_ptr;
typedef __attribute__((address_space(3))) v4i* as3_v4i_ptr;

// nfloats must be a multiple of 4; src 16B-aligned (hipMalloc allocations are).
__device__ __forceinline__ void stage_to_lds(const float* __restrict__ src,
                                             float* __restrict__ dst_lds,
                                             int nfloats) {
#ifdef ASYNC_LDS
  for (int i = (int)threadIdx.x * 4; i < nfloats; i += (int)blockDim.x * 4) {
    __builtin_amdgcn_global_load_async_to_lds_b128(
        (as1_v4i_ptr)(void*)(src + i),
        (as3_v4i_ptr)(void*)(dst_lds + i),
        0, 0);
  }
  __builtin_amdgcn_s_wait_asynccnt(0);
  __syncthreads();
#else
  for (int i = (int)threadIdx.x; i < nfloats; i += (int)blockDim.x)
    dst_lds[i] = src[i];
  __syncthreads();
#endif
}

// ---------------------------------------------------------------------------
// K0: zero scatter grid + output scalar (must happen every call)
// ---------------------------------------------------------------------------
__global__ void k_init(float* __restrict__ grid, float* __restrict__ out, int nf) {
  int i = blockIdx.x * blockDim.x + threadIdx.x;
  if (i < nf) grid[i] = 0.f;
  if (i == 0) out[0] = 0.f;
}

// ---------------------------------------------------------------------------
// K1: per-particle MLP forward + scatter-add into grid [B,GZ,GZ,2]
// ---------------------------------------------------------------------------
__global__ void __launch_bounds__(256)
k_scatter(const float* __restrict__ pts,
          const float* __restrict__ W1, const float* __restrict__ B1,
          const float* __restrict__ W2, const float* __restrict__ B2,
          const float* __restrict__ W3, const float* __restrict__ B3,
          float* __restrict__ grid) {
  int t = blockIdx.x * blockDim.x + threadIdx.x;
  if (t >= NB * NPTS) return;
  int b = t / NPTS;

  float w1[8], b1[4], w2[16], b2[4], w3[8], b3[2];
#pragma unroll
  for (int i = 0; i < 8;  ++i) w1[i] = W1[i];
#pragma unroll
  for (int i = 0; i < 4;  ++i) b1[i] = B1[i];
#pragma unroll
  for (int i = 0; i < 16; ++i) w2[i] = W2[i];
#pragma unroll
  for (int i = 0; i < 4;  ++i) b2[i] = B2[i];
#pragma unroll
  for (int i = 0; i < 8;  ++i) w3[i] = W3[i];
#pragma unroll
  for (int i = 0; i < 2;  ++i) b3[i] = B3[i];

  float p0 = pts[2 * t], p1 = pts[2 * t + 1];
  float c0 = floorf(p0), c1 = floorf(p1);
  int   i0 = (int)c0,    i1 = (int)c1;

#pragma unroll
  for (int cc = 0; cc < 4; ++cc) {
    float a = (float)(cc >> 1), d = (float)(cc & 1);
    float f0 = p0 - c0 - a, f1 = p1 - c1 - d;
    float h[4], g[4];
#pragma unroll
    for (int j = 0; j < 4; ++j) h[j] = elu_f(b1[j] + f0 * w1[j] + f1 * w1[4 + j]);
#pragma unroll
    for (int j = 0; j < 4; ++j)
      g[j] = elu_f(b2[j] + h[0]*w2[j] + h[1]*w2[4+j] + h[2]*w2[8+j] + h[3]*w2[12+j]);
    float v0 = b3[0], v1 = b3[1];
#pragma unroll
    for (int j = 0; j < 4; ++j) { v0 += g[j] * w3[2*j]; v1 += g[j] * w3[2*j + 1]; }

    int iy = clampi(i0 + (cc >> 1), 0, GZ - 1);
    int ix = clampi(i1 + (cc & 1),  0, GZ - 1);
    float* cell = grid + (((size_t)b * GZ + iy) * GZ + ix) * 2;
    atomicAdd(cell,     v0);
    atomicAdd(cell + 1, v1);
  }
}

// ---------------------------------------------------------------------------
// K2: h3[c,y,x] = adjoint(cw4) applied to all-ones cotangent (border sums)
// ---------------------------------------------------------------------------
__global__ void k_h3(const float* __restrict__ cw4, float* __restrict__ h3) {
  int t = blockIdx.x * blockDim.x + threadIdx.x;
  if (t >= NCH * HZ * HZ) return;
  int c = t / (HZ * HZ);
  int r = t % (HZ * HZ);
  int y = r / HZ, x = r % HZ;
  float acc = 0.f;
  for (int ky = 0; ky < 5; ++ky) {
    int yy = y - ky + 2; if (yy < 0 || yy >= HZ) continue;
    for (int kx = 0; kx < 5; ++kx) {
      int xx = x - kx + 2; if (xx < 0 || xx >= HZ) continue;
      for (int o = 0; o < NCH; ++o)
        acc += cw4[((o * NCH + c) * 5 + ky) * 5 + kx];
    }
  }
  h3[t] = acc;
}

// ---------------------------------------------------------------------------
// K3: adjoint 5x5 conv (stride 1, pad 2), 8->8 channels, via WMMA f32 16x16x4.
//   gout[c,y,x] = sum_{o,ky,kx} w[o,c,ky,kx] * gprev[o, y-ky+2, x-kx+2]
// GEMM view: M=16 x-pixels, N=16 (8 used, = c), K=8*25=200 -> 50 wmma steps,
// fully unrolled so the k->(o,ky,kx) decode folds to constants.
// Weights staged to LDS once per block via GLOBAL_LOAD_ASYNC_TO_LDS_B128.
// One wave per (y, x-chunk-of-16); 4 waves per block.
// ---------------------------------------------------------------------------
__global__ void __launch_bounds__(128)
k_adjconv_wmma(const float* __restrict__ gprev, const float* __restrict__ w,
               float* __restrict__ gout) {
  __shared__ float lw[NCH * NCH * 25];      // 1600 floats = 6.4 KB
  stage_to_lds(w, lw, NCH * NCH * 25);

  int wave = blockIdx.x * (blockDim.x >> 5) + (threadIdx.x >> 5);
  int L = threadIdx.x & 31;
  int y  = wave / (HZ / 16);
  int x0 = (wave % (HZ / 16)) * 16;
  int m  = L & 15;            // A: pixel row; B: output column (channel c)
  int kh = (L >> 4) << 1;     // lanes 0-15 -> K+{0,1}; lanes 16-31 -> K+{2,3}
  bool valid_c = (m < NCH);

  v8f C = {};
#pragma unroll
  for (int t = 0; t < 50; ++t) {
    v2f A, Bf;
#pragma unroll
    for (int q = 0; q < 2; ++q) {
      int k  = 4 * t + kh + q;
      int o  = k / 25, rr = k % 25;
      int ky = rr / 5, kx = rr % 5;
      int yy = y - ky + 2;
      int xx = (x0 + m) - kx + 2;
      float av = 0.f;
      if (yy >= 0 && yy < HZ && xx >= 0 && xx < HZ)
        av = gprev[(o * HZ + yy) * HZ + xx];
      float bv = valid_c ? lw[((o * NCH + m) * 5 + ky) * 5 + kx] : 0.f;
      if (q == 0) { A.x = av; Bf.x = bv; } else { A.y = av; Bf.y = bv; }
    }
    C = __builtin_amdgcn_wmma_f32_16x16x4_f32(false, A, false, Bf,
                                              (short)0, C, false, false);
  }
  if (valid_c) {
    int mb = (L >> 4) * 8;
#pragma unroll
    for (int r2 = 0; r2 < 8; ++r2)
      gout[(m * HZ + y) * HZ + x0 + mb + r2] = C[r2];
  }
}

// ---------------------------------------------------------------------------
// K4: conv1 forward (stride 2, pad 2, 2->8 ch) fused with g1 = h1 * elu'(y1),
// via WMMA f32 16x16x4. K = 2*25 = 50 -> 13 steps (2 zero-padded), unrolled.
// ---------------------------------------------------------------------------
__global__ void __launch_bounds__(128)
k_conv1g_wmma(const float* __restrict__ grid, const float* __restrict__ cw1,
              const float* __restrict__ cb1, const float* __restrict__ h1,
              float* __restrict__ g1) {
  __shared__ float lw[NCH * 2 * 25];        // 400 floats
  stage_to_lds(cw1, lw, NCH * 2 * 25);

  int wave = blockIdx.x * (blockDim.x >> 5) + (threadIdx.x >> 5);
  int L = threadIdx.x & 31;
  int b   = wave / (HZ * (HZ / 16));
  int rem = wave % (HZ * (HZ / 16));
  int y   = rem / (HZ / 16);
  int x0  = (rem % (HZ / 16)) * 16;
  int m   = L & 15;
  int kh  = (L >> 4) << 1;
  bool valid_o = (m < NCH);

  v8f C = {};
#pragma unroll
  for (int t = 0; t < 13; ++t) {
    v2f A, Bf;
#pragma unroll
    for (int q = 0; q < 2; ++q) {
      int k = 4 * t + kh + q;
      float av = 0.f, bv = 0.f;
      if (k < 50) {
        int c  = k / 25, rr = k % 25;
        int ky = rr / 5, kx = rr % 5;
        int Y = 2 * y + ky - 2;
        int X = 2 * (x0 + m) + kx - 2;
        if (Y >= 0 && Y < GZ && X >= 0 && X < GZ)
          av = grid[(((size_t)b * GZ + Y) * GZ + X) * 2 + c];
        if (valid_o)
          bv = lw[((m * 2 + c) * 5 + ky) * 5 + kx];
      }
      if (q == 0) { A.x = av; Bf.x = bv; } else { A.y = av; Bf.y = bv; }
    }
    C = __builtin_amdgcn_wmma_f32_16x16x4_f32(false, A, false, Bf,
                                              (short)0, C, false, false);
  }
  if (valid_o) {
    int mb = (L >> 4) * 8;
    float bias = cb1[m];
#pragma unroll
    for (int r2 = 0; r2 < 8; ++r2) {
      int x = x0 + mb + r2;
      float y1 = C[r2] + bias;
      g1[(((size_t)b * NCH + m) * HZ + y) * HZ + x] =
          h1[(m * HZ + y) * HZ + x] * elup_f(y1);
    }
  }
}

// ---------------------------------------------------------------------------
// K5: adjoint of stride-2 conv1 -> g_grid stored as [B,GZ,GZ,2]
// ---------------------------------------------------------------------------
__global__ void k_adjconv1(const float* __restrict__ g1,
                           const float* __restrict__ cw1,
                           float* __restrict__ gg) {
  int t = blockIdx.x * blockDim.x + threadIdx.x;
  if (t >= NB * 2 * GZ * GZ) return;
  int b = t / (2 * GZ * GZ);
  int r = t % (2 * GZ * GZ);
  int c = r / (GZ * GZ); r %= (GZ * GZ);
  int Y = r / GZ, X = r % GZ;

  float acc = 0.f;
#pragma unroll
  for (int ky = 0; ky < 5; ++ky) {
    int ty = Y + 2 - ky;
    if (ty & 1) continue;
    int yy = ty >> 1;
    if (yy < 0 || yy >= HZ) continue;
#pragma unroll
    for (int kx = 0; kx < 5; ++kx) {
      int tx = X + 2 - kx;
      if (tx & 1) continue;
      int xx = tx >> 1;
      if (xx < 0 || xx >= HZ) continue;
#pragma unroll
      for (int o = 0; o < NCH; ++o)
        acc += cw1[((o * 2 + c) * 5 + ky) * 5 + kx] *
               g1[(((size_t)b * NCH + o) * HZ + yy) * HZ + xx];
    }
  }
  gg[(((size_t)b * GZ + Y) * GZ + X) * 2 + c] = acc;
}

// ---------------------------------------------------------------------------
// K6: per-particle gather + MLP backward + squared residual, block-reduced
// ---------------------------------------------------------------------------
__global__ void __launch_bounds__(256)
k_loss(const float* __restrict__ pts, const float* __restrict__ goals,
       const float* __restrict__ W1, const float* __restrict__ B1,
       const float* __restrict__ W2, const float* __restrict__ B2,
       const float* __restrict__ W3, const float* __restrict__ B3,
       const float* __restrict__ gg, float* __restrict__ out) {
  int t = blockIdx.x * blockDim.x + threadIdx.x;
  float acc = 0.f;
  if (t < NB * NPTS) {
    float w1[8], b1[4], w2[16], b2[4], w3[8];
#pragma unroll
    for (int i = 0; i < 8;  ++i) w1[i] = W1[i];
#pragma unroll
    for (int i = 0; i < 4;  ++i) b1[i] = B1[i];
#pragma unroll
    for (int i = 0; i < 16; ++i) w2[i] = W2[i];
#pragma unroll
    for (int i = 0; i < 4;  ++i) b2[i] = B2[i];
#pragma unroll
    for (int i = 0; i < 8;  ++i) w3[i] = W3[i];
    (void)B3; // b3 has zero gradient contribution to points

    int b = t / NPTS;
    float p0 = pts[2 * t], p1 = pts[2 * t + 1];
    float c0 = floorf(p0), c1 = floorf(p1);
    int   i0 = (int)c0,    i1 = (int)c1;
    float dp0 = 0.f, dp1 = 0.f;

#pragma unroll
    for (int cc = 0; cc < 4; ++cc) {
      float a = (float)(cc >> 1), d = (float)(cc & 1);
      float f0 = p0 - c0 - a, f1 = p1 - c1 - d;
      float pre1[4], a1[4];
#pragma unroll
      for (int j = 0; j < 4; ++j) {
        pre1[j] = b1[j] + f0 * w1[j] + f1 * w1[4 + j];
        a1[j] = elu_f(pre1[j]);
      }
      float pre2[4];
#pragma unroll
      for (int j = 0; j < 4; ++j)
        pre2[j] = b2[j] + a1[0]*w2[j] + a1[1]*w2[4+j] + a1[2]*w2[8+j] + a1[3]*w2[12+j];

      int iy = clampi(i0 + (cc >> 1), 0, GZ - 1);
      int ix = clampi(i1 + (cc & 1),  0, GZ - 1);
      const float* cell = gg + (((size_t)b * GZ + iy) * GZ + ix) * 2;
      float u0 = cell[0], u1 = cell[1];

      float dq[4];
#pragma unroll
      for (int j = 0; j < 4; ++j)
        dq[j] = (w3[2*j] * u0 + w3[2*j + 1] * u1) * elup_f(pre2[j]);
      float dr[4];
#pragma unroll
      for (int i = 0; i < 4; ++i)
        dr[i] = (w2[i*4]*dq[0] + w2[i*4+1]*dq[1] + w2[i*4+2]*dq[2] + w2[i*4+3]*dq[3]) *
                elup_f(pre1[i]);
      dp0 += w1[0]*dr[0] + w1[1]*dr[1] + w1[2]*dr[2] + w1[3]*dr[3];
      dp1 += w1[4]*dr[0] + w1[5]*dr[1] + w1[6]*dr[2] + w1[7]*dr[3];
    }

    const float inv = 1.f / (float)NPTS;
    float r0 = (dp0 - goals[2 * t])     * inv;
    float r1 = (dp1 - goals[2 * t + 1]) * inv;
    acc = r0 * r0 + r1 * r1;
  }

  __shared__ float sdata[256];
  sdata[threadIdx.x] = acc;
  __syncthreads();
  for (int s = 128; s > 0; s >>= 1) {
    if (threadIdx.x < s) sdata[threadIdx.x] += sdata[threadIdx.x + s];
    __syncthreads();
  }
  if (threadIdx.x == 0) atomicAdd(out, sdata[0]);
}

// ---------------------------------------------------------------------------
// Launch
// ---------------------------------------------------------------------------
extern "C" void kernel_launch(void* const* d_in, const int* in_sizes, int n_in,
                              void* d_out, int out_size, void* d_ws, size_t ws_size,
                              hipStream_t stream) {
  const float* pts   = (const float*)d_in[0];
  const float* goals = (const float*)d_in[1];
  const float* W1    = (const float*)d_in[2];
  const float* B1    = (const float*)d_in[3];
  const float* W2    = (const float*)d_in[4];
  const float* B2    = (const float*)d_in[5];
  const float* W3    = (const float*)d_in[6];
  const float* B3    = (const float*)d_in[7];
  const float* cw1   = (const float*)d_in[8];
  const float* cb1   = (const float*)d_in[9];
  const float* cw2   = (const float*)d_in[10];
  // cb2, cb3, cb4 (d_in[11,13,15]) contribute constants to E -> zero grad
  const float* cw3   = (const float*)d_in[12];
  const float* cw4   = (const float*)d_in[14];
  float* out = (float*)d_out;

  // workspace layout (bytes)
  constexpr size_t SZ_GRID = (size_t)NB * GZ * GZ * 2 * 4;   // 4,718,592
  constexpr size_t SZ_H    = (size_t)NCH * HZ * HZ * 4;      // 1,179,648
  constexpr size_t SZ_G1   = (size_t)NB * NCH * HZ * HZ * 4; // 4,718,592
  char* ws = (char*)d_ws;
  float* grid = (float*)(ws);
  float* h3   = (float*)(ws + SZ_GRID);
  float* h2   = (float*)(ws + SZ_GRID + SZ_H);
  float* h1   = (float*)(ws + SZ_GRID + 2 * SZ_H);
  float* g1   = (float*)(ws + SZ_GRID + 3 * SZ_H);
  float* gg   = (float*)(ws + SZ_GRID + 3 * SZ_H + SZ_G1);

  const int nf = NB * GZ * GZ * 2;
  k_init<<<(nf + 255) / 256, 256, 0, stream>>>(grid, out, nf);

  k_scatter<<<(NB * NPTS + 255) / 256, 256, 0, stream>>>(
      pts, W1, B1, W2, B2, W3, B3, grid);

  k_h3<<<(NCH * HZ * HZ + 255) / 256, 256, 0, stream>>>(cw4, h3);

  // h2 = adj(cw3, h3); h1 = adj(cw2, h2) — 192*12 = 2304 waves, 4 waves/block
  k_adjconv_wmma<<<(HZ * (HZ / 16)) / 4, 128, 0, stream>>>(h3, cw3, h2);
  k_adjconv_wmma<<<(HZ * (HZ / 16)) / 4, 128, 0, stream>>>(h2, cw2, h1);

  // g1 = h1 * elu'(conv1(grid)) — NB*192*12 = 9216 waves
  k_conv1g_wmma<<<(NB * HZ * (HZ / 16)) / 4, 128, 0, stream>>>(
      grid, cw1, cb1, h1, g1);

  k_adjconv1<<<(NB * 2 * GZ * GZ + 255) / 256, 256, 0, stream>>>(g1, cw1, gg);

  k_loss<<<(NB * NPTS + 255) / 256, 256, 0, stream>>>(
      pts, goals, W1, B1, W2, B2, W3, B3, gg, out);

  (void)in_sizes; (void)n_in; (void)out_size; (void)ws_size;
}